// FKineLinked_23416161697926
// MI455X (gfx1250) — compile-verified
//
#include <hip/hip_runtime.h>
#include <math.h>

// FKine forward: per-joint MLP (3->128, 128->128 x2, 128->12) + chained 4x4.
// fp32 WMMA 16x16x4 path: D = A(16x4) x B(4x16) + C, fp32 throughout.
// 8 waves / 256 threads per block, 128 samples per block; per-wave LDS scratch.

typedef __attribute__((ext_vector_type(2))) float v2f;
typedef __attribute__((ext_vector_type(8))) float v8f;

#define JN 7
#define HD 128
#define WAVES 8
#define TILE_M 16
#define SAMPLES_PER_BLOCK (WAVES * TILE_M)  // 128
#define LROW 132                            // padded fp32 row stride (conflict-free b64 LDS loads)

__global__ __launch_bounds__(WAVES * 32) void fkine_wmma_kernel(
    const float* __restrict__ q,   // [N, 7]
    const float* __restrict__ W0,  // [7, 128, 3]
    const float* __restrict__ b0,  // [7, 128]
    const float* __restrict__ Wh,  // [7, 2, 128, 128]
    const float* __restrict__ bh,  // [7, 2, 128]
    const float* __restrict__ Wf,  // [7, 12, 128]
    const float* __restrict__ bf,  // [7, 12]
    float* __restrict__ y_out,     // [N, 3, 7]
    float* __restrict__ t_out,     // [N, 7, 4, 4]
    int nsamples)
{
    __shared__ float sh[WAVES * TILE_M * LROW];     // per-wave activation scratch [16][132]
    __shared__ float stop[SAMPLES_PER_BLOCK * 12];  // final-layer outputs (3x4 per sample)

    const int tid   = threadIdx.x;
    const int wave  = tid >> 5;
    const int lane  = tid & 31;
    const int nlane = lane & 15;   // N (or M for A-frag) index within tile
    const int hi    = lane >> 4;   // lane half selects K pair {0,1} vs {2,3}

    const int block_s0 = blockIdx.x * SAMPLES_PER_BLOCK;
    const int wave_s0  = block_s0 + wave * TILE_M;

    float* hact = &sh[wave * TILE_M * LROW];

    // chain state t_prev = identity (per-sample; threads 0..127 own samples)
    float tp[16];
#pragma unroll
    for (int i = 0; i < 16; ++i) tp[i] = ((i % 5) == 0) ? 1.0f : 0.0f;

    // clamped sample row for this lane's A-fragment (keeps EXEC full for WMMA)
    int srow = wave_s0 + nlane;
    if (srow > nsamples - 1) srow = nsamples - 1;

    for (int j = 0; j < JN; ++j) {
        // hint the L2 path for this joint's big weight blocks
        __builtin_prefetch(Wh + ((size_t)j * 2) * HD * HD, 0, 1);

        // ---------------- layer 0: qf[16x4(3 used)] @ W0^T -> h[16][128], relu ----------------
        {
            float theta = q[(size_t)srow * JN + j];
            float c = cosf(theta);
            float s = sinf(theta);
            // A-frag (fp32 16x4): lanes0-15: K=0,1 ; lanes16-31: K=2,3 ; K pad (=3) is zero
            v2f a0;
            a0.x = hi ? s : theta;
            a0.y = hi ? 0.0f : c;
            const float* W0j = W0 + (size_t)j * HD * 3;
            const float* b0j = b0 + (size_t)j * HD;
#pragma unroll
            for (int nt = 0; nt < 8; ++nt) {
                int n = nt * 16 + nlane;
                const float* wrow = W0j + (size_t)n * 3;  // torch layout [out, in]
                // unconditional loads + register selects: no exec-masked load paths
                float w0v = wrow[0];
                float w1v = wrow[1];
                float w2v = wrow[2];
                v2f bfr;
                bfr.x = hi ? w2v : w0v;
                bfr.y = hi ? 0.0f : w1v;
                v8f acc = {};
                acc = __builtin_amdgcn_wmma_f32_16x16x4_f32(
                    false, a0, false, bfr, (short)0, acc, false, false);
                float bias = b0j[n];
#pragma unroll
                for (int r = 0; r < 8; ++r) {
                    float v = acc[r] + bias;
                    hact[(r + hi * 8) * LROW + n] = fmaxf(v, 0.0f);
                }
            }
        }
        // scratch is wave-private; LDS ops are in-order per wave -> no barrier needed

        // ---------------- hidden layers: h = relu(h @ Wh[l]^T + bh[l]), K=128 ----------------
        for (int l = 0; l < 2; ++l) {
            const float* W  = Wh + (size_t)(j * 2 + l) * HD * HD;
            const float* bb = bh + (size_t)(j * 2 + l) * HD;
            v8f acc[8];
#pragma unroll
            for (int nt = 0; nt < 8; ++nt) acc[nt] = (v8f){};
            for (int k = 0; k < 32; ++k) {
                int kb = k * 4 + hi * 2;
                v2f a = *(const v2f*)&hact[nlane * LROW + kb];  // ds_load_b64, conflict-free
#pragma unroll
                for (int nt = 0; nt < 8; ++nt) {
                    const float* wp = W + (size_t)(nt * 16 + nlane) * HD + kb;
                    v2f bw = *(const v2f*)wp;  // global b64, WGP$/L2-resident, shared by 8 waves
                    acc[nt] = __builtin_amdgcn_wmma_f32_16x16x4_f32(
                        false, a, false, bw, (short)0, acc[nt], false, false);
                }
            }
            // all loads issued before stores; DS in-order per wave -> in-place update is safe
#pragma unroll
            for (int nt = 0; nt < 8; ++nt) {
                int n = nt * 16 + nlane;
                float bias = bb[n];
#pragma unroll
                for (int r = 0; r < 8; ++r) {
                    float v = acc[nt][r] + bias;
                    hact[(r + hi * 8) * LROW + n] = fmaxf(v, 0.0f);
                }
            }
        }

        // ---------------- final layer: top[16x12] = h @ Wf^T + bf (no relu) ----------------
        {
            const float* W  = Wf + (size_t)j * 12 * HD;
            const float* bb = bf + (size_t)j * 12;
            const int nvalid = (nlane < 12);
            const float* wrow = W + (size_t)(nvalid ? nlane : 0) * HD;  // clamp: load stays unconditional
            v8f acc = {};
            for (int k = 0; k < 32; ++k) {
                int kb = k * 4 + hi * 2;
                v2f a = *(const v2f*)&hact[nlane * LROW + kb];
                v2f bw = *(const v2f*)(wrow + kb);
                bw.x = nvalid ? bw.x : 0.0f;
                bw.y = nvalid ? bw.y : 0.0f;
                acc = __builtin_amdgcn_wmma_f32_16x16x4_f32(
                    false, a, false, bw, (short)0, acc, false, false);
            }
            if (nvalid) {
                float bias = bb[nlane];
#pragma unroll
                for (int r = 0; r < 8; ++r)
                    stop[(wave * 16 + r + hi * 8) * 12 + nlane] = acc[r] + bias;
            }
        }
        __syncthreads();

        // ---------------- chain phase: t_new = t_local @ t_prev ----------------
        if (tid < SAMPLES_PER_BLOCK) {
            int gs = block_s0 + tid;
            if (gs < nsamples) {
                float tl[16];
#pragma unroll
                for (int i = 0; i < 12; ++i) tl[i] = stop[tid * 12 + i];
                tl[12] = 0.0f; tl[13] = 0.0f; tl[14] = 0.0f; tl[15] = 1.0f;
                float tn[16];
#pragma unroll
                for (int r = 0; r < 4; ++r)
#pragma unroll
                    for (int c2 = 0; c2 < 4; ++c2) {
                        float s = 0.0f;
#pragma unroll
                        for (int kk = 0; kk < 4; ++kk)
                            s = fmaf(tl[r * 4 + kk], tp[kk * 4 + c2], s);
                        tn[r * 4 + c2] = s;
                    }
                // t output = LOCAL transform, [N][7][4][4]
                float4* tptr = (float4*)(t_out + ((size_t)gs * JN + j) * 16);
                tptr[0] = make_float4(tl[0], tl[1], tl[2], tl[3]);
                tptr[1] = make_float4(tl[4], tl[5], tl[6], tl[7]);
                tptr[2] = make_float4(tl[8], tl[9], tl[10], tl[11]);
                tptr[3] = make_float4(tl[12], tl[13], tl[14], tl[15]);
                // y = cumulative translation, [N][3][7]
                y_out[(size_t)gs * 21 + 0 * JN + j] = tn[3];
                y_out[(size_t)gs * 21 + 1 * JN + j] = tn[7];
                y_out[(size_t)gs * 21 + 2 * JN + j] = tn[11];
#pragma unroll
                for (int i = 0; i < 16; ++i) tp[i] = tn[i];
            }
        }
        __syncthreads();
    }
}

extern "C" void kernel_launch(void* const* d_in, const int* in_sizes, int n_in,
                              void* d_out, int out_size, void* d_ws, size_t ws_size,
                              hipStream_t stream) {
    const float* q  = (const float*)d_in[0];
    const float* W0 = (const float*)d_in[1];
    const float* b0 = (const float*)d_in[2];
    const float* Wh = (const float*)d_in[3];
    const float* bh = (const float*)d_in[4];
    const float* Wf = (const float*)d_in[5];
    const float* bf = (const float*)d_in[6];

    int nsamples = in_sizes[0] / JN;
    float* y_out = (float*)d_out;
    float* t_out = y_out + (size_t)nsamples * 3 * JN;

    int grid = (nsamples + SAMPLES_PER_BLOCK - 1) / SAMPLES_PER_BLOCK;
    hipLaunchKernelGGL(fkine_wmma_kernel, dim3(grid), dim3(WAVES * 32), 0, stream,
                       q, W0, b0, Wh, bh, Wf, bf, y_out, t_out, nsamples);
}